// Mask_Model_68779606278183
// MI455X (gfx1250) — compile-verified
//
#include <hip/hip_runtime.h>

#define NU   50000
#define NI   30000
#define NNZE 1600000
#define EMB  128
// TAU = 0.5 -> multiply by 2
#define INV_TAU 2.0f

typedef __attribute__((ext_vector_type(2))) float v2f;
typedef __attribute__((ext_vector_type(8))) float v8f;

// ---- monotone float<->uint mapping so unsigned atomicMax == float max ----
__device__ __forceinline__ unsigned fkey(float f) {
    unsigned u = __float_as_uint(f);
    return (u & 0x80000000u) ? ~u : (u | 0x80000000u);
}
__device__ __forceinline__ float fdecode(unsigned k) {
    unsigned u = (k & 0x80000000u) ? (k & 0x7FFFFFFFu) : ~k;
    return __uint_as_float(u);
}

// ---------------------------------------------------------------------------
// Out[m, n] = sum_k X[m,k] * W[n,k] + bias[n]   (X: [M,128], W: [128,128])
// One wave computes a 16x16 tile via V_WMMA_F32_16X16X4_F32, K looped 128/4.
// Block = 256 threads = 8 waves = the 8 column tiles of N=128.
// M is an exact multiple of 16, so no predication (EXEC all ones for WMMA).
// ---------------------------------------------------------------------------
__global__ __launch_bounds__(256) void proj_gemm(
    const float* __restrict__ X, const float* __restrict__ W,
    const float* __restrict__ bias, float* __restrict__ Out)
{
    const int wave = threadIdx.x >> 5;
    const int lane = threadIdx.x & 31;
    const int half = lane >> 4;    // selects K pair (A) / K pair (B) / M half (D)
    const int lid  = lane & 15;
    const int mBase = blockIdx.x << 4;
    const int nBase = wave << 4;

    // A fragment source: row m = mBase+lid, elems (k+2*half, k+2*half+1)
    const float* __restrict__ arow = X + (size_t)(mBase + lid) * EMB;
    // B fragment source: B[k][n] = W[n][k]; row n = nBase+lid, contiguous in k
    const float* __restrict__ brow = W + (size_t)(nBase + lid) * EMB;

    v8f acc = {0.f, 0.f, 0.f, 0.f, 0.f, 0.f, 0.f, 0.f};
#pragma unroll
    for (int k = 0; k < EMB; k += 4) {
        const int kk = k + half * 2;
        const float2 af = *(const float2*)(arow + kk);
        const float2 bf = *(const float2*)(brow + kk);
        v2f a; a.x = af.x; a.y = af.y;
        v2f b; b.x = bf.x; b.y = bf.y;
        acc = __builtin_amdgcn_wmma_f32_16x16x4_f32(
            /*neg_a=*/false, a, /*neg_b=*/false, b,
            /*c_mod=*/(short)0, acc, /*reuse_a=*/false, /*reuse_b=*/false);
    }

    // D layout: lane covers column N = nBase+lid; VGPR v is row mBase + v + 8*half
    const float bv = bias[nBase + lid];
    float* __restrict__ op = Out + (size_t)(mBase + half * 8) * EMB + nBase + lid;
#pragma unroll
    for (int v = 0; v < 8; ++v)
        op[(size_t)v * EMB] = acc[v] + bv;
}

// ---------------------------------------------------------------------------
// One wave32 per edge: coalesced 512B gathers of Qu[r], Ki[c], Qi[c], Ku[r],
// lane-parallel dot + xor-shuffle reduction. Writes logits into d_out and
// accumulates per-segment running max via ordered-uint atomicMax.
// ---------------------------------------------------------------------------
__global__ __launch_bounds__(256) void edge_logits(
    const float* __restrict__ Qu, const float* __restrict__ Ki,
    const float* __restrict__ Qi, const float* __restrict__ Ku,
    const float* __restrict__ adj, const float* __restrict__ gui,
    const float* __restrict__ giu, const int* __restrict__ rows,
    const int* __restrict__ cols, float* __restrict__ out,
    unsigned* __restrict__ umax, unsigned* __restrict__ imax)
{
    const int e    = (int)((blockIdx.x * blockDim.x + threadIdx.x) >> 5);
    const int lane = threadIdx.x & 31;
    if (e >= NNZE) return;
    const int r = rows[e], c = cols[e];

    const float4 qu = ((const float4*)(Qu + (size_t)r * EMB))[lane];
    const float4 ki = ((const float4*)(Ki + (size_t)c * EMB))[lane];
    const float4 qi = ((const float4*)(Qi + (size_t)c * EMB))[lane];
    const float4 ku = ((const float4*)(Ku + (size_t)r * EMB))[lane];

    float pui = qu.x * ki.x + qu.y * ki.y + qu.z * ki.z + qu.w * ki.w;
    float piu = qi.x * ku.x + qi.y * ku.y + qi.z * ku.z + qi.w * ku.w;
#pragma unroll
    for (int off = 16; off > 0; off >>= 1) {
        pui += __shfl_xor(pui, off, 32);
        piu += __shfl_xor(piu, off, 32);
    }

    if (lane == 0) {
        const float a   = adj[e];
        const float zui = (a * pui - __logf(-__logf(gui[e]))) * INV_TAU;
        const float ziu = (a * piu - __logf(-__logf(giu[e]))) * INV_TAU;
        out[e]        = zui;
        out[NNZE + e] = ziu;
        atomicMax(&umax[r], fkey(zui));
        atomicMax(&imax[c], fkey(ziu));
    }
}

// exp(z - segMax) in place + per-segment sum via atomicAdd
__global__ __launch_bounds__(256) void edge_exp(
    const int* __restrict__ rows, const int* __restrict__ cols,
    float* __restrict__ out,
    const unsigned* __restrict__ umax, const unsigned* __restrict__ imax,
    float* __restrict__ usum, float* __restrict__ isum)
{
    const int e = blockIdx.x * blockDim.x + threadIdx.x;
    if (e >= NNZE) return;
    const int r = rows[e], c = cols[e];
    const float eui = __expf(out[e]        - fdecode(umax[r]));
    const float eiu = __expf(out[NNZE + e] - fdecode(imax[c]));
    out[e]        = eui;
    out[NNZE + e] = eiu;
    atomicAdd(&usum[r], eui);
    atomicAdd(&isum[c], eiu);
}

// normalize by segment sum
__global__ __launch_bounds__(256) void edge_norm(
    const int* __restrict__ rows, const int* __restrict__ cols,
    float* __restrict__ out,
    const float* __restrict__ usum, const float* __restrict__ isum)
{
    const int e = blockIdx.x * blockDim.x + threadIdx.x;
    if (e >= NNZE) return;
    out[e]        /= usum[rows[e]];
    out[NNZE + e] /= isum[cols[e]];
}

extern "C" void kernel_launch(void* const* d_in, const int* in_sizes, int n_in,
                              void* d_out, int out_size, void* d_ws, size_t ws_size,
                              hipStream_t stream) {
    const float* user_embed = (const float*)d_in[0];
    const float* item_embed = (const float*)d_in[1];
    const float* Wq  = (const float*)d_in[2];
    const float* bq  = (const float*)d_in[3];
    const float* Wk  = (const float*)d_in[4];
    const float* bk  = (const float*)d_in[5];
    const float* adj = (const float*)d_in[6];
    const float* gui = (const float*)d_in[7];
    const float* giu = (const float*)d_in[8];
    const int*   rows = (const int*)d_in[9];
    const int*   cols = (const int*)d_in[10];
    float* out = (float*)d_out;

    // workspace layout: 4 projection matrices + per-segment max/sum accumulators
    float*    Qu   = (float*)d_ws;
    float*    Ku   = Qu + (size_t)NU * EMB;
    float*    Qi   = Ku + (size_t)NU * EMB;
    float*    Ki   = Qi + (size_t)NI * EMB;
    unsigned* umax = (unsigned*)(Ki + (size_t)NI * EMB);
    float*    usum = (float*)(umax + NU);
    unsigned* imax = (unsigned*)(usum + NU);
    float*    isum = (float*)(imax + NI);

    // reset accumulators every call (umax/imax: key 0 == -inf sentinel; sums: 0.0f)
    hipMemsetAsync(umax, 0, (size_t)(2 * NU + 2 * NI) * sizeof(float), stream);

    // 4 WMMA projection GEMMs (M exact multiples of 16; N=128 = 8 waves/block)
    proj_gemm<<<NU / 16, 256, 0, stream>>>(user_embed, Wq, bq, Qu);
    proj_gemm<<<NU / 16, 256, 0, stream>>>(user_embed, Wk, bk, Ku);
    proj_gemm<<<NI / 16, 256, 0, stream>>>(item_embed, Wq, bq, Qi);
    proj_gemm<<<NI / 16, 256, 0, stream>>>(item_embed, Wk, bk, Ki);

    // edge phase: 1 wave per edge -> NNZ*32 threads
    edge_logits<<<(NNZE * 32) / 256, 256, 0, stream>>>(
        Qu, Ki, Qi, Ku, adj, gui, giu, rows, cols, out, umax, imax);
    edge_exp<<<NNZE / 256, 256, 0, stream>>>(rows, cols, out, umax, imax, usum, isum);
    edge_norm<<<NNZE / 256, 256, 0, stream>>>(rows, cols, out, usum, isum);
}